// ModeLayer_79474074845702
// MI455X (gfx1250) — compile-verified
//
#include <hip/hip_runtime.h>
#include <stdint.h>

// ---------------------------------------------------------------------------
// Per-row mode (B=65536 rows, N=512 int32 each, C=1000 classes) -> one-hot f32.
// Memory-bound kernel (~384 MB traffic -> ~16.5us @ 23.3 TB/s). CDNA5 path:
// TDM async global->LDS load (tensor_load_to_lds + s_wait_tensorcnt), LDS
// atomics for histogram/first-pos, non-temporal b128 stores for the 256 MB
// write-once one-hot output (output > 192 MB L2, so NT avoids L2 thrash).
// ---------------------------------------------------------------------------

typedef __attribute__((ext_vector_type(4))) unsigned int v4u;
typedef __attribute__((ext_vector_type(8))) int          v8i;
typedef __attribute__((ext_vector_type(4))) int          v4i;
typedef __attribute__((ext_vector_type(4))) float        v4f;

#define N_ELEM 512
#define N_CLS  1000
#define C_PAD  1024

__global__ __launch_bounds__(256) void mode_onehot_kernel(
    const int* __restrict__ in, float* __restrict__ out, int B) {
  __shared__ int          stage[N_ELEM];   // 2 KB staged row (TDM destination)
  __shared__ int          cnt[C_PAD];      // per-class count
  __shared__ int          fpos[C_PAD];     // per-class first-occurrence position
  __shared__ unsigned int bestkey;

  const int t   = threadIdx.x;
  const int row = blockIdx.x;
  if (row >= B) return;

  // --- init histogram state (runs concurrently with TDM issue) ---
#pragma unroll
  for (int i = 0; i < 4; ++i) {
    cnt[t + i * 256]  = 0;
    fpos[t + i * 256] = N_ELEM;            // "absent" sentinel
  }
  if (t == 0) bestkey = 0u;

  // --- wave 0 issues TDM: copy row (512 x i32 = 2 KB) global -> LDS ---
  if (t < 32) {
    uint64_t ga  = (uint64_t)(uintptr_t)(in + (size_t)row * N_ELEM);
    uint32_t lds = (uint32_t)(uintptr_t)(&stage[0]);   // LDS byte offset (addr[31:0])

    // D# group 0 (128b): count=1 | lds_addr | global_addr[56:0] | type=2
    v4u g0;
    g0[0] = 1u;                                          // count=1, user descriptor
    g0[1] = lds;                                         // lds_addr
    g0[2] = (uint32_t)ga;                                // global_addr[31:0]
    g0[3] = (uint32_t)((ga >> 32) & 0x01FFFFFFu)         // global_addr[56:32]
            | (2u << 30);                                // type = 2 ("image")

    // D# group 1 (256b): mask=0, data_size=4B, tensor_dim0=512, tile_dim0=512
    v8i g1;
    g1[0] = 0x00020000;          // workgroup_mask=0 (not in cluster), data_size=2 (4B)
    g1[1] = (N_ELEM << 16);      // tensor_dim0[15:0] = 512 (bits 63:48)
    g1[2] = (1 << 16);           // tensor_dim0[31:16]=0 ; tensor_dim1[15:0]=1
    g1[3] = (N_ELEM << 16);      // tensor_dim1[31:16]=0 ; tile_dim0 = 512
    g1[4] = 1;                   // tile_dim1 = 1, tile_dim2 = 0
    g1[5] = N_ELEM;              // tensor_dim0_stride[31:0] = 512
    g1[6] = 0;                   // stride hi / dim1_stride lo
    g1[7] = 0;

    v4i gz4 = {0, 0, 0, 0};                  // groups 2/3 unused (<=2D tensor)
    v8i gz8 = {0, 0, 0, 0, 0, 0, 0, 0};      // 5th descriptor group (clang-23 arity)

    __builtin_amdgcn_tensor_load_to_lds(g0, g1, gz4, gz4, gz8, 0);
    __builtin_amdgcn_s_wait_tensorcnt(0);   // drain TENSORcnt on issuing wave
  }
  __syncthreads();                          // publish staged row to all 8 waves

  // --- histogram + first-occurrence via LDS atomics (2 elements/thread) ---
#pragma unroll
  for (int i = 0; i < 2; ++i) {
    int p = t + i * 256;
    int v = stage[p];
    atomicAdd(&cnt[v], 1);                  // ds_add_u32
    atomicMin(&fpos[v], p);                 // ds_min_i32
  }
  __syncthreads();

  // --- argmax with TF tie-break: score = (count<<10) | (N - first_pos) ---
  // first_pos is unique per present class => max key unique; absent classes
  // (and padding 1000..1023) have key < 1024 and never win.
  unsigned int local = 0;
#pragma unroll
  for (int i = 0; i < 4; ++i) {
    int c = t + i * 256;
    unsigned int score = ((unsigned int)cnt[c] << 10)
                       | (unsigned int)(N_ELEM - fpos[c]);
    unsigned int key = (score << 10) | (unsigned int)c;   // < 2^30
    local = (key > local) ? key : local;
  }
  atomicMax(&bestkey, local);               // ds_max_u32
  __syncthreads();

  const int mode = (int)(bestkey & 1023u);

  // --- one-hot output: 250 non-temporal float4 (b128) stores, coalesced.
  // Output stream (256 MB/call) exceeds the 192 MB L2 -> NT keeps L2 clean.
  if (t < 250) {
    v4f q = {0.f, 0.f, 0.f, 0.f};
    if ((mode >> 2) == t) q[mode & 3] = 1.0f;
    v4f* orow = (v4f*)(out + (size_t)row * N_CLS);        // 4000B stride, 16-aligned
    __builtin_nontemporal_store(q, orow + t);             // th:TH_STORE_NT
  }
}

extern "C" void kernel_launch(void* const* d_in, const int* in_sizes, int n_in,
                              void* d_out, int out_size, void* d_ws, size_t ws_size,
                              hipStream_t stream) {
  const int* in  = (const int*)d_in[0];
  float*     out = (float*)d_out;
  const int  B   = in_sizes[0] / N_ELEM;    // 65536
  hipLaunchKernelGGL(mode_onehot_kernel, dim3(B), dim3(256), 0, stream,
                     in, out, B);
}